// QuantumModel_70050916598212
// MI455X (gfx1250) — compile-verified
//
#include <hip/hip_runtime.h>
#include <math.h>

typedef __attribute__((ext_vector_type(2))) float v2f;
typedef __attribute__((ext_vector_type(8))) float v8f;

#define WAVES_PER_BLOCK 8
#define TILES_PER_WAVE 4
#define SAMPLES_PER_BLOCK (WAVES_PER_BLOCK * TILES_PER_WAVE * 16)

// lane <-> lane^16 half-wave swap: ds_swizzle_b32 SWAPX16 (xor=0x10,and=0x1f)
__device__ __forceinline__ float swap_half(float v) {
  return __int_as_float(__builtin_amdgcn_ds_swizzle(__float_as_int(v), 0x401F));
}

// ---------------------------------------------------------------------------
// Setup: build the fixed 16x16 unitary M = R(w1)*C*R(w0)*C (column per lane),
// the fused readout matrix G[2][16] = W @ Z-signs, and copy bias.
// Qubit i corresponds to bit (3-i) of the flattened state index (qubit 0 = MSB,
// matching the reference's (B,2,2,2,2) layout).
// ---------------------------------------------------------------------------
__global__ void qsim_setup(const float* __restrict__ w,     // (2,4)
                           const float* __restrict__ W,     // (2,4)
                           const float* __restrict__ bias,  // (2,)
                           float* __restrict__ Mmat,        // 256 floats
                           float* __restrict__ G,           // 32 floats
                           float* __restrict__ bcopy)       // 2 floats
{
  const int t = threadIdx.x;
  if (t < 16) {
    float v[16];
#pragma unroll
    for (int k = 0; k < 16; ++k) v[k] = 0.f;
    v[t] = 1.f;
    for (int l = 0; l < 2; ++l) {
      // CNOT ring: i -> (i+1)%4, applied in order i=0..3
#pragma unroll
      for (int i = 0; i < 4; ++i) {
        const int cm = 1 << (3 - i);
        const int tm = 1 << (3 - ((i + 1) & 3));
        float nv[16];
#pragma unroll
        for (int k = 0; k < 16; ++k) nv[k] = v[(k & cm) ? (k ^ tm) : k];
#pragma unroll
        for (int k = 0; k < 16; ++k) v[k] = nv[k];
      }
      // RY(weights[l][i]) on each qubit
#pragma unroll
      for (int i = 0; i < 4; ++i) {
        const float half = 0.5f * w[l * 4 + i];
        const float c = cosf(half), s = sinf(half);
        const int qm = 1 << (3 - i);
#pragma unroll
        for (int k = 0; k < 16; ++k) {
          if (!(k & qm)) {
            const float a0 = v[k], a1 = v[k | qm];
            v[k]      = c * a0 - s * a1;
            v[k | qm] = s * a0 + c * a1;
          }
        }
      }
    }
    // column t of M
#pragma unroll
    for (int r = 0; r < 16; ++r) Mmat[r * 16 + t] = v[r];
  }
  if (t == 16) {
#pragma unroll
    for (int j = 0; j < 2; ++j)
      for (int m = 0; m < 16; ++m) {
        float g = 0.f;
#pragma unroll
        for (int i = 0; i < 4; ++i)
          g += W[j * 4 + i] * (((m >> (3 - i)) & 1) ? -1.f : 1.f);
        G[j * 16 + m] = g;
      }
    bcopy[0] = bias[0];
    bcopy[1] = bias[1];
  }
}

// ---------------------------------------------------------------------------
// Main streaming kernel: per wave-tile of 16 samples,
//   psi(16 x 16 samples) = M(16x16) * S0(16 x 16)  via 4x V_WMMA_F32_16X16X4_F32
//   out = G * psi^2 + b
// ---------------------------------------------------------------------------
__global__ void qsim_main(const float* __restrict__ x,
                          const float* __restrict__ Mmat,
                          const float* __restrict__ G,
                          const float* __restrict__ bvec,
                          float* __restrict__ out,
                          int batch)
{
  const int lane = threadIdx.x & 31;
  const int wave = threadIdx.x >> 5;
  const int hi   = lane >> 4;   // half-wave: selects K (A/B) and M-offset (D)
  const int nl   = lane & 15;   // A row / B column / sample-in-tile

  // A fragments: chunk q covers K = 4q..4q+3 of M; lane holds A[nl, 2hi(+1)]
  v2f a[4];
#pragma unroll
  for (int q = 0; q < 4; ++q) {
    const int col = 4 * q + 2 * hi;
    a[q].x = Mmat[nl * 16 + col];
    a[q].y = Mmat[nl * 16 + col + 1];
  }
  // readout rows for this lane's D half (m = 8*hi + v)
  float g0[8], g1[8];
#pragma unroll
  for (int v = 0; v < 8; ++v) {
    g0[v] = G[8 * hi + v];
    g1[v] = G[16 + 8 * hi + v];
  }
  const float b0 = bvec[0], b1 = bvec[1];
  // phase trick: t2[hi] = hi ? sin(h2) : cos(h2) = sin(h2 + (hi ? 0 : pi/2))
  const float ph2 = hi ? 0.f : 1.57079632679f;

  const long s0id = ((long)blockIdx.x * WAVES_PER_BLOCK + wave) *
                        (TILES_PER_WAVE * 16L) + nl;
  const float* __restrict__ xp = x + s0id * 4;   // lanes 16-31 mirror 0-15
  float* __restrict__ op       = out + s0id * 2;
  const long rem = (long)batch - s0id;           // tile t in-bounds iff t*16 < rem

#pragma unroll
  for (int t = 0; t < TILES_PER_WAVE; ++t) {
    const bool inb = (t * 16) < rem;
    float4 xs = make_float4(0.f, 0.f, 0.f, 0.f);
    if (inb) xs = *(const float4*)(xp + t * 64);  // global_load_b128 (+imm offs)

    // half-angle sin/cos: qubits 0,1,3 need both; qubit 2 needs one per half
    float c0, s0, c1, s1, c3, s3;
    __sincosf(0.5f * xs.x, &s0, &c0);
    __sincosf(0.5f * xs.y, &s1, &c1);
    __sincosf(0.5f * xs.w, &s3, &c3);
    const float t2v = __sinf(fmaf(0.5f, xs.z, ph2));

    // s0-state amplitudes: amp[k] = t0[k>>3]*t1[(k>>2)&1]*t2[(k>>1)&1]*t3[k&1]
    float p01[4];
    p01[0] = c0 * c1; p01[1] = c0 * s1; p01[2] = s0 * c1; p01[3] = s0 * s1;
    const float p23a = t2v * c3;
    const float p23b = t2v * s3;

    // psi = M * s0 : 4 chained f32 WMMAs over K
    v8f acc = {};
#pragma unroll
    for (int q = 0; q < 4; ++q) {
      v2f bf;
      bf.x = p01[q] * p23a;  // S0[4q + 2hi    , n]
      bf.y = p01[q] * p23b;  // S0[4q + 2hi + 1, n]
      acc = __builtin_amdgcn_wmma_f32_16x16x4_f32(
          false, a[q], false, bf, (short)0, acc, false, false);
    }

    // readout: f_j = sum_v G[j, 8hi+v] * psi^2 ; combine halves via ds_swizzle
    float f0 = 0.f, f1 = 0.f;
#pragma unroll
    for (int v = 0; v < 8; ++v) {
      const float p = acc[v] * acc[v];
      f0 = fmaf(g0[v], p, f0);
      f1 = fmaf(g1[v], p, f1);
    }
    f0 += swap_half(f0);
    f1 += swap_half(f1);

    if (hi == 0 && inb) {
      v2f o;
      o.x = f0 + b0;
      o.y = f1 + b1;
      *(v2f*)(op + t * 32) = o;  // global_store_b64, coalesced (+imm offs)
    }
  }
}

// ---------------------------------------------------------------------------
extern "C" void kernel_launch(void* const* d_in, const int* in_sizes, int n_in,
                              void* d_out, int out_size, void* d_ws, size_t ws_size,
                              hipStream_t stream)
{
  const float* x  = (const float*)d_in[0];  // (B,4)
  const float* w  = (const float*)d_in[1];  // (2,4)
  const float* W  = (const float*)d_in[2];  // (2,4)
  const float* b  = (const float*)d_in[3];  // (2,)
  float* out = (float*)d_out;               // (B,2)
  const int batch = in_sizes[0] / 4;

  float* Mmat = (float*)d_ws;   // 256
  float* G    = Mmat + 256;     // 32
  float* bcpy = G + 32;         // 2

  hipLaunchKernelGGL(qsim_setup, dim3(1), dim3(32), 0, stream,
                     w, W, b, Mmat, G, bcpy);

  const int grid = (batch + SAMPLES_PER_BLOCK - 1) / SAMPLES_PER_BLOCK;
  hipLaunchKernelGGL(qsim_main, dim3(grid), dim3(WAVES_PER_BLOCK * 32), 0, stream,
                     x, Mmat, G, bcpy, out, batch);
}